// LowRankLayer_dilation_75960791597921
// MI455X (gfx1250) — compile-verified
//
#include <hip/hip_runtime.h>

typedef float v2f __attribute__((ext_vector_type(2)));
typedef float v8f __attribute__((ext_vector_type(8)));

static constexpr int BATCH = 4, CH = 64, Hs = 128, Wsp = 128;
static constexpr int HW   = Hs * Wsp;        // 16384 pixels per batch image
static constexpr int NPIX = BATCH * HW;      // 65536 total pixels
static constexpr float EPSV = 1e-6f;

// ---------------------------------------------------------------------------
// Kernel A: h[(b*HW+n)*64 + o] = relu( sum_c Whead[o][c] * x[b,c,n] )
// One wave = one 16(o) x 16(pixel) WMMA tile, K=64 in steps of 4 (fp32 WMMA).
// Output transposed to (pixel, channel) so the NMF kernel loads coalesce.
// ---------------------------------------------------------------------------
__global__ __launch_bounds__(128) void head_gemm(const float* __restrict__ x,
                                                 const float* __restrict__ Wh,
                                                 float* __restrict__ h) {
  const int lane = threadIdx.x & 31;
  const int wave = threadIdx.x >> 5;
  const int blk  = blockIdx.x;           // 0 .. 4*1024-1
  const int b    = blk >> 10;
  const int n0   = (blk & 1023) << 4;    // pixel tile base
  const int m0   = wave << 4;            // out-channel tile base
  const int ln   = lane & 15;
  const int hi   = lane >> 4;            // 0: K={0,1}, 1: K={2,3}

  const float* xb = x + (size_t)b * CH * HW;
  v8f acc = {};
#pragma unroll
  for (int k0 = 0; k0 < 64; k0 += 4) {
    const int ka = k0 + 2 * hi;
    v2f a, bm;
    a.x  = Wh[(m0 + ln) * 64 + ka];                 // A: M = m0+ln, K = ka
    a.y  = Wh[(m0 + ln) * 64 + ka + 1];
    bm.x = xb[(size_t)ka * HW + n0 + ln];           // B: K = ka, N = n0+ln
    bm.y = xb[(size_t)(ka + 1) * HW + n0 + ln];
    acc = __builtin_amdgcn_wmma_f32_16x16x4_f32(false, a, false, bm,
                                                (short)0, acc, false, false);
  }
  // D: lane -> pixel n0+ln ; VGPR r -> channel m0 + r + 8*hi (contiguous in r)
  float* hp = h + ((size_t)(b * HW + n0 + ln)) * CH + m0 + 8 * hi;
#pragma unroll
  for (int r = 0; r < 8; ++r) hp[r] = fmaxf(acc[r], 0.0f);
}

// ---------------------------------------------------------------------------
// Kernel B: one wave per pixel. Rank-collapsed exact NMF update.
// Lane l holds channels l and l+32; 9 dilated replicate-padded taps in regs.
// 19 channel-reductions batched through one shfl_xor tree.
// ---------------------------------------------------------------------------
__global__ __launch_bounds__(256) void nmf_pixel(const float* __restrict__ h,
                                                 float* __restrict__ t) {
  const int lane = threadIdx.x & 31;
  const int wave = threadIdx.x >> 5;
  const int pix  = blockIdx.x * 8 + wave;   // 0 .. NPIX-1
  const int b  = pix >> 14;
  const int n  = pix & (HW - 1);
  const int py = n >> 7, px = n & 127;

  float X0[9], X1[9];
#pragma unroll
  for (int kh = 0; kh < 3; ++kh) {
    int yy = py + 2 * kh - 2; yy = yy < 0 ? 0 : (yy > 127 ? 127 : yy);
#pragma unroll
    for (int kw = 0; kw < 3; ++kw) {
      int xx = px + 2 * kw - 2; xx = xx < 0 ? 0 : (xx > 127 ? 127 : xx);
      const float* hp = h + ((size_t)(b * HW + yy * 128 + xx)) * CH;
      X0[kh * 3 + kw] = hp[lane];
      X1[kh * 3 + kw] = hp[lane + 32];
    }
  }
  float u0 = 0.f, u1 = 0.f;
#pragma unroll
  for (int k = 0; k < 9; ++k) { u0 += X0[k]; u1 += X1[k]; }
  u0 *= (1.f / 9.f); u1 *= (1.f / 9.f);

  // partials: p[0..8] = n_k, p[9..17] = 64*v_k, p[18] = uu
  float p[19];
#pragma unroll
  for (int k = 0; k < 9; ++k) p[k]     = u0 * X0[k] + u1 * X1[k];
#pragma unroll
  for (int k = 0; k < 9; ++k) p[9 + k] = X0[k] + X1[k];
  p[18] = u0 * u0 + u1 * u1;

#pragma unroll
  for (int off = 16; off; off >>= 1) {
#pragma unroll
    for (int i = 0; i < 19; ++i) p[i] += __shfl_xor(p[i], off, 32);
  }

  const float uu = p[18];
  float w[9], ww = 0.f;
#pragma unroll
  for (int k = 0; k < 9; ++k) {
    const float vk = p[9 + k] * (1.f / 64.f);
    const float wk = vk * p[k] / (3.f * uu * vk + EPSV);
    w[k] = wk; ww += wk * wk;
  }
  float m0 = 0.f, m1 = 0.f;
#pragma unroll
  for (int k = 0; k < 9; ++k) { m0 += X0[k] * w[k]; m1 += X1[k] * w[k]; }
  const float U0 = u0 * m0 / (3.f * u0 * ww + EPSV);
  const float U1 = u1 * m1 / (3.f * u1 * ww + EPSV);
  const float w4 = w[4];

  float* tp = t + (size_t)pix * CH;
  tp[lane]      = 3.f * U0 * w4;
  tp[lane + 32] = 3.f * U1 * w4;
}

// ---------------------------------------------------------------------------
// Kernel C: out[b,c,n] = sum_o Wtail[c][o] * t[(b,n),o] + x[b,c,n]
// Flipped GEMM: A = 16(pixel) x 4(o) tile of t, B = Wtail^T -> D = pixel x c,
// which stores row-contiguous into the (b,c,h,w) output, fused residual.
// ---------------------------------------------------------------------------
__global__ __launch_bounds__(128) void tail_gemm(const float* __restrict__ t,
                                                 const float* __restrict__ Wt,
                                                 const float* __restrict__ x,
                                                 float* __restrict__ out) {
  const int lane = threadIdx.x & 31;
  const int wave = threadIdx.x >> 5;
  const int blk  = blockIdx.x;
  const int b    = blk >> 10;
  const int p0   = (blk & 1023) << 4;    // pixel tile base (M)
  const int n0   = wave << 4;            // out-channel tile base (N)
  const int ln   = lane & 15;
  const int hi   = lane >> 4;

  const float* tb = t + (size_t)b * HW * CH;
  v8f acc = {};
#pragma unroll
  for (int k0 = 0; k0 < 64; k0 += 4) {
    const int ka = k0 + 2 * hi;
    v2f a, bm;
    a.x  = tb[(size_t)(p0 + ln) * CH + ka];         // A: M = pixel, K = o
    a.y  = tb[(size_t)(p0 + ln) * CH + ka + 1];
    bm.x = Wt[(n0 + ln) * 64 + ka];                 // B[o][c] = Wtail[c][o]
    bm.y = Wt[(n0 + ln) * 64 + ka + 1];
    acc = __builtin_amdgcn_wmma_f32_16x16x4_f32(false, a, false, bm,
                                                (short)0, acc, false, false);
  }
  const int c = n0 + ln;                 // lane -> channel (N)
  const size_t base = ((size_t)(b * CH + c)) * HW + p0 + 8 * hi;
  const float* xr = x + base;
  float* op = out + base;
#pragma unroll
  for (int r = 0; r < 8; ++r) op[r] = acc[r] + xr[r];
}

extern "C" void kernel_launch(void* const* d_in, const int* in_sizes, int n_in,
                              void* d_out, int out_size, void* d_ws, size_t ws_size,
                              hipStream_t stream) {
  (void)in_sizes; (void)n_in; (void)out_size; (void)ws_size;
  const float* x  = (const float*)d_in[0];
  const float* Wh = (const float*)d_in[1];   // (out_c, in_c)
  const float* Wt = (const float*)d_in[2];   // (in_c, out_c)
  float* out = (float*)d_out;

  float* h = (float*)d_ws;                       // NPIX*64 floats (16.8 MB)
  float* t = h + (size_t)NPIX * CH;              // NPIX*64 floats (16.8 MB)

  head_gemm<<<dim3(BATCH * 1024), dim3(128), 0, stream>>>(x, Wh, h);
  nmf_pixel<<<dim3(NPIX / 8),     dim3(256), 0, stream>>>(h, t);
  tail_gemm<<<dim3(BATCH * 1024), dim3(128), 0, stream>>>(t, Wt, x, out);
}